// NegativeSamplingLoss_26070451486829
// MI455X (gfx1250) — compile-verified
//
#include <hip/hip_runtime.h>
#include <hip/hip_bf16.h>
#include <stdint.h>

// ---------------- feature probes (compile-safe on either toolchain) ---------
#ifndef __has_builtin
#define __has_builtin(x) 0
#endif

#if __has_builtin(__builtin_amdgcn_wmma_f32_16x16x4_f32)
#define HAVE_WMMA4 1
#else
#define HAVE_WMMA4 0
#endif

#if __has_builtin(__builtin_amdgcn_global_load_async_to_lds_b128) && \
    __has_builtin(__builtin_amdgcn_s_wait_asynccnt)
#define HAVE_ASYNC 1
#else
#define HAVE_ASYNC 0
#endif

typedef __attribute__((ext_vector_type(2))) float v2f;
typedef __attribute__((ext_vector_type(8))) float v8f;

#if HAVE_ASYNC
// Exact pointee type from the compiler diagnostic:
//   __attribute__((__vector_size__(4 * sizeof(int)))) int  (i.e. <4 x i32>)
// with AS1 (global / "__device__") source and AS3 (LDS / "__shared__") dest.
typedef int v4i_ __attribute__((__vector_size__(16)));
typedef __attribute__((address_space(1))) v4i_ as1_v4i;
typedef __attribute__((address_space(3))) v4i_ as3_v4i;
#endif

// ---------------- problem constants (from reference) ------------------------
#define BATCH   1024
#define NCOL    100000
#define TPB     256
#define NWAVES  (TPB / 32)
#define CHUNK   (TPB * 4)                     // 1024 floats per chunk
#define NCH     ((NCOL + CHUNK - 1) / CHUNK)  // 98 chunks
#define TOPK    8
#define LOG2E   1.44269504088896340736f

// ---------------- wave32 sum via WMMA (D = A x ones + 0) --------------------
// A is 16x4 fp32: lane m holds K0/K1, lane m+16 holds K2/K3. With a.x=x, a.y=0
// every D[m][n] = x_m + x_{m+16}; c[0..7] on lane 0 sums half the wave, lane 16
// the other half -> one wmma + one shfl = full 32-lane reduction.
__device__ __forceinline__ float wave_sum32(float x) {
#if HAVE_WMMA4
  v2f a; a[0] = x;   a[1] = 0.0f;
  v2f b; b[0] = 1.0f; b[1] = 1.0f;
  v8f c = {};
  c = __builtin_amdgcn_wmma_f32_16x16x4_f32(false, a, false, b, (short)0, c,
                                            false, false);
  float s = c[0] + c[1] + c[2] + c[3] + c[4] + c[5] + c[6] + c[7];
  s += __shfl(s, 16, 32);   // lane0: first half + second half = total
  return s;                 // valid on lanes 0..15 (we use lane 0)
#else
  #pragma unroll
  for (int o = 16; o > 0; o >>= 1) x += __shfl_xor(x, o, 32);
  return x;
#endif
}

// ---------------- kernel 1: per-row streaming reduction ---------------------
__global__ __launch_bounds__(TPB)
void nsl_row_kernel(const float* __restrict__ pred,
                    const long long* __restrict__ targets,
                    float* __restrict__ ws_focal,
                    float* __restrict__ ws_rank) {
  const int row = blockIdx.x;
  const int tid = threadIdx.x;
  const int lane = tid & 31;
  const int wv = tid >> 5;
  const int tgt = (int)targets[row];
  const float* __restrict__ rowp = pred + (size_t)row * NCOL;

  __shared__ __align__(16) float buf[2][CHUNK];   // double buffer (8 KB)
  __shared__ float wm[NWAVES], wsum[NWAVES];
  __shared__ float cand_v[TPB * TOPK];
  __shared__ int   cand_i[TPB * TOPK];

  // per-thread online softmax state + top-8 (excluding target column)
  float m = -INFINITY, s = 0.0f;
  float topv[TOPK];
  int   topi[TOPK];
  #pragma unroll
  for (int k = 0; k < TOPK; ++k) { topv[k] = -INFINITY; topi[k] = -1; }

  auto upd = [&](float v, int gi) {
    if (v > m) { s *= exp2f((m - v) * LOG2E); m = v; }   // rare rescale
    s += exp2f((v - m) * LOG2E);
    if (gi != tgt && v > topv[TOPK - 1]) {
      float cv = v; int ci = gi;
      #pragma unroll
      for (int k = 0; k < TOPK; ++k) {
        if (cv > topv[k]) {
          float tv = topv[k]; int ti = topi[k];
          topv[k] = cv; topi[k] = ci;
          cv = tv; ci = ti;
        }
      }
    }
  };

#if HAVE_ASYNC
  // ---- async double-buffered stream: DMA chunk c+1 while computing chunk c.
  #define ISSUE_CHUNK(c_) do {                                                 \
      int b4_ = (c_) * TPB + tid;                 /* float4 index in row */     \
      if (b4_ < NCOL / 4) {                                                    \
        const float* gsrc_ = rowp + (size_t)b4_ * 4;                           \
        __builtin_amdgcn_global_load_async_to_lds_b128(                        \
            (as1_v4i*)(uintptr_t)gsrc_,                                        \
            (as3_v4i*)(uint32_t)(uintptr_t)&buf[(c_) & 1][tid * 4], 0, 0);     \
      }                                                                        \
    } while (0)

  ISSUE_CHUNK(0);
  for (int c = 0; c < NCH; ++c) {
    __builtin_amdgcn_s_wait_asynccnt(0);   // this wave's DMA for chunk c done
    __syncthreads();                        // all waves' DMA landed in LDS
    if (c + 1 < NCH) ISSUE_CHUNK(c + 1);    // prefetch into the other buffer
    int gbase = c * CHUNK + tid * 4;
    if (gbase < NCOL) {
      float4 v = *reinterpret_cast<const float4*>(&buf[c & 1][tid * 4]);
      upd(v.x, gbase); upd(v.y, gbase + 1); upd(v.z, gbase + 2); upd(v.w, gbase + 3);
    }
  }
  #undef ISSUE_CHUNK
#else
  // ---- fallback: direct coalesced b128 loads from HBM
  for (int c = 0; c < NCH; ++c) {
    int gbase = c * CHUNK + tid * 4;
    if (gbase < NCOL) {
      float4 v = *reinterpret_cast<const float4*>(rowp + gbase);
      upd(v.x, gbase); upd(v.y, gbase + 1); upd(v.z, gbase + 2); upd(v.w, gbase + 3);
    }
  }
#endif

  // ---- wave-level log-sum-exp merge (shfl butterfly) ----
  #pragma unroll
  for (int o = 16; o > 0; o >>= 1) {
    float om_ = __shfl_xor(m, o, 32);
    float os_ = __shfl_xor(s, o, 32);
    float nm = fmaxf(m, om_);
    s = s * exp2f((m - nm) * LOG2E) + os_ * exp2f((om_ - nm) * LOG2E);
    m = nm;
  }
  if (lane == 0) { wm[wv] = m; wsum[wv] = s; }

  // ---- dump per-thread top-8 candidates ----
  #pragma unroll
  for (int k = 0; k < TOPK; ++k) {
    cand_v[tid * TOPK + k] = topv[k];
    cand_i[tid * TOPK + k] = topi[k];
  }
  __syncthreads();

  if (tid == 0) {
    // merge 8 wave (m,s) pairs
    float M = wm[0], S = wsum[0];
    for (int i = 1; i < NWAVES; ++i) {
      float nm = fmaxf(M, wm[i]);
      S = S * exp2f((M - nm) * LOG2E) + wsum[i] * exp2f((wm[i] - nm) * LOG2E);
      M = nm;
    }
    float pos = rowp[tgt];
    float lse = M + log2f(S) / LOG2E;
    float ce  = lse - pos;
    float p_t = exp2f((pos - lse) * LOG2E);
    float om1 = fmaxf(1.0f - p_t, 0.0f);
    float focal = 0.5f * om1 * sqrtf(om1) * ce;        // alpha*(1-p)^1.5*CE

    // block top-5: ranks 0..2 = hard negatives, 3..4 = "popular" negatives
    float t5[5] = {-INFINITY, -INFINITY, -INFINITY, -INFINITY, -INFINITY};
    for (int i = 0; i < TPB * TOPK; ++i) {
      float cv = cand_v[i];
      if (cv > t5[4]) {
        #pragma unroll
        for (int k = 0; k < 5; ++k) {
          if (cv > t5[k]) { float tv = t5[k]; t5[k] = cv; cv = tv; }
        }
      }
    }
    float rsum = 0.0f;
    #pragma unroll
    for (int k = 0; k < 5; ++k) {
      float l = 1.0f - (pos - t5[k]);
      rsum += fmaxf(l, 0.0f);
    }
    // 5 random negatives via hash PRNG, gathered straight from HBM
    unsigned seed = 0x9E3779B9u * (unsigned)(row + 1);
    for (int j = 0; j < 5; ++j) {
      seed = seed * 1664525u + 1013904223u + (unsigned)j;
      unsigned h = seed ^ (seed >> 16);
      h *= 0x85EBCA6Bu; h ^= h >> 13;
      int ri = (int)(h % (unsigned)NCOL);
      if (ri == tgt) ri = (ri + 1) % NCOL;
      float l = 1.0f - (pos - rowp[ri]);
      rsum += fmaxf(l, 0.0f);
    }
    ws_focal[row] = focal;
    ws_rank[row]  = rsum * 0.1f;   // mean over 10 negatives
  }
}

// ---------------- kernel 2: final reduction (one wave, WMMA tree) -----------
__global__ __launch_bounds__(32)
void nsl_final_kernel(const float* __restrict__ ws_focal,
                      const float* __restrict__ ws_rank,
                      float* __restrict__ out) {
  const int lane = threadIdx.x;
  float pf = 0.0f, pr = 0.0f;
  for (int i = lane; i < BATCH; i += 32) {
    pf += ws_focal[i];
    pr += ws_rank[i];
  }
  float sf = wave_sum32(pf);
  float sr = wave_sum32(pr);
  if (lane == 0) {
    float focal = sf * (1.0f / BATCH);
    float rank  = sr * (1.0f / BATCH);
    out[0] = 0.7f * focal + 0.3f * rank;  // total
    out[1] = focal;
    out[2] = rank;
  }
}

// ---------------- launch ----------------------------------------------------
extern "C" void kernel_launch(void* const* d_in, const int* in_sizes, int n_in,
                              void* d_out, int out_size, void* d_ws, size_t ws_size,
                              hipStream_t stream) {
  (void)in_sizes; (void)n_in; (void)out_size; (void)ws_size;
  const float*     pred = (const float*)d_in[0];
  const long long* tgt  = (const long long*)d_in[1];
  float* ws_focal = (float*)d_ws;
  float* ws_rank  = ws_focal + BATCH;

  nsl_row_kernel<<<BATCH, TPB, 0, stream>>>(pred, tgt, ws_focal, ws_rank);
  nsl_final_kernel<<<1, 32, 0, stream>>>(ws_focal, ws_rank, (float*)d_out);
}